// LightGCNLayer_87866440942267
// MI455X (gfx1250) — compile-verified
//
#include <hip/hip_runtime.h>

typedef __attribute__((ext_vector_type(2))) float v2f;
typedef __attribute__((ext_vector_type(4))) float v4f;
typedef __attribute__((ext_vector_type(8))) float v8f;

#define GN 16384            // nodes (adj is GN x GN)
#define GD 64               // feature dim
#define BK 64               // K-slab depth staged in LDS
#define BROWS 64            // output rows per workgroup (4 waves * 16)
#define NTHREADS 128        // 4 waves (wave32)
#define LDS_STRIDE 72       // 64 + 8 pad -> half-wave groups hit disjoint banks
#define PDROP 0.1f
#define PKEEP_INV (1.0f / 0.9f)
#define PF_SLABS 2          // prefetch distance (K-slabs ahead) for adj stream

__global__ __launch_bounds__(NTHREADS) void lightgcn_wmma_f32(
    const float* __restrict__ x,
    const float* __restrict__ adj,
    const float* __restrict__ u,
    float* __restrict__ out)
{
    __shared__ __align__(16) float lds[2][BK * LDS_STRIDE];

    const int tid   = threadIdx.x;
    const int wave  = tid >> 5;
    const int lane  = tid & 31;
    const int lhalf = lane >> 4;       // 0 or 1
    const int llow  = lane & 15;
    const int koff  = lhalf * 2;       // A/B K-offset for upper half-wave

    const int    mWave = blockIdx.x * BROWS + wave * 16;
    const size_t rowA  = (size_t)(mWave + llow) * (size_t)GN;

    // Stage one 64x64 slab of x_drop into LDS (dropout fused on the fly).
    auto load_slab = [&](int s, int buf) {
        const size_t base = (size_t)s * (size_t)(BK * GD);
        const float* xs = x + base;
        const float* us = u + base;
        #pragma unroll
        for (int c = 0; c < 8; ++c) {
            const int linear = c * 512 + tid * 4;     // 4096 floats / 128 thr
            const int row = linear >> 6;
            const int col = linear & 63;
            v4f xv = *(const v4f*)(xs + linear);
            v4f uv = *(const v4f*)(us + linear);
            v4f d;
            d.x = (uv.x >= PDROP) ? xv.x * PKEEP_INV : 0.0f;
            d.y = (uv.y >= PDROP) ? xv.y * PKEEP_INV : 0.0f;
            d.z = (uv.z >= PDROP) ? xv.z * PKEEP_INV : 0.0f;
            d.w = (uv.w >= PDROP) ? xv.w * PKEEP_INV : 0.0f;
            *(v4f*)(&lds[buf][row * LDS_STRIDE + col]) = d;
        }
    };

    v8f acc[4];
    {
        v8f z = {};
        #pragma unroll
        for (int t = 0; t < 4; ++t) acc[t] = z;
    }

    load_slab(0, 0);
    __syncthreads();

    const int NS = GN / BK;            // 256 K-slabs
    for (int s = 0; s < NS; ++s) {
        const int    cur   = s & 1;
        const size_t kBase = (size_t)s * (size_t)BK;

        // Prefetch this lane's adj stream PF_SLABS ahead (global_prefetch_b8).
        if (s + PF_SLABS < NS)
            __builtin_prefetch(adj + rowA + kBase + (size_t)(PF_SLABS * BK)
                                   + (size_t)(lane * 2), 0, 1);

        // Preload this slab's A fragments (global_load_b64 per k-step).
        // Layout: lanes 0-15 hold (K=k, K=k+1), lanes 16-31 hold (K=k+2, K=k+3).
        v2f aReg[16];
        const float* pA = adj + rowA + kBase + (size_t)koff;
        #pragma unroll
        for (int kk = 0; kk < 16; ++kk)
            aReg[kk] = *(const v2f*)(pA + kk * 4);

        // Overlap: stage next slab while A loads are in flight.
        if (s + 1 < NS) load_slab(s + 1, cur ^ 1);

        const float* B = &lds[cur][0];
        #pragma unroll
        for (int kk = 0; kk < 16; ++kk) {
            const int r0 = (kk * 4 + koff) * LDS_STRIDE + llow;
            #pragma unroll
            for (int nt = 0; nt < 4; ++nt) {
                v2f b;
                b.x = B[r0 + nt * 16];
                b.y = B[r0 + LDS_STRIDE + nt * 16];
                acc[nt] = __builtin_amdgcn_wmma_f32_16x16x4_f32(
                    /*neg_a=*/false, aReg[kk],
                    /*neg_b=*/false, b,
                    /*c_mod=*/(short)0, acc[nt],
                    /*reuse_a=*/false, /*reuse_b=*/false);
            }
        }
        __syncthreads();
    }

    // C/D layout: VGPR r -> (M=r, N=lane) for lanes 0-15, (M=r+8, N=lane-16).
    float* po = out + (size_t)(mWave + lhalf * 8) * (size_t)GD + llow;
    #pragma unroll
    for (int nt = 0; nt < 4; ++nt) {
        #pragma unroll
        for (int r = 0; r < 8; ++r)
            po[(size_t)r * GD + nt * 16] = acc[nt][r];
    }
}

extern "C" void kernel_launch(void* const* d_in, const int* in_sizes, int n_in,
                              void* d_out, int out_size, void* d_ws, size_t ws_size,
                              hipStream_t stream) {
    (void)in_sizes; (void)n_in; (void)out_size; (void)d_ws; (void)ws_size;
    const float* x   = (const float*)d_in[0];
    const float* adj = (const float*)d_in[1];
    const float* u   = (const float*)d_in[2];
    float* out       = (float*)d_out;

    dim3 grid(GN / BROWS);   // 256 workgroups
    dim3 block(NTHREADS);    // 4 waves of 32
    lightgcn_wmma_f32<<<grid, block, 0, stream>>>(x, adj, u, out);
}